// MSGDC_73194832659261
// MI455X (gfx1250) — compile-verified
//
#include <hip/hip_runtime.h>
#include <stdint.h>

typedef float v2f __attribute__((ext_vector_type(2)));
typedef float v8f __attribute__((ext_vector_type(8)));

#define C_ALL   192
#define NC      8                    // channels staged per chunk
#define NCHUNK  (C_ALL / NC)         // 24
#define TILE_H  14                   // 4 output rows + 2*5 halo
#define TILE_W  38                   // 28 output cols + 2*5 halo
#define TILE_CH (TILE_H * TILE_W)    // 532 floats per channel
#define TILE_ELEMS (NC * TILE_CH)    // 4256
#define ILOAD   17                   // ceil(4256 / 256): async issues per thread per chunk
#define SPOS    112                  // 4 x 28 spatial positions per workgroup

__device__ __forceinline__ uint32_t lds_off(const void* p) {
    // addrspacecast(shared->flat) puts the LDS byte offset in the low 32 bits
    return (uint32_t)(uintptr_t)p;
}

__global__ __launch_bounds__(256) void msgdc_fused(
    const float* __restrict__ x,      const float* __restrict__ bias,
    const float* __restrict__ conv_w, const float* __restrict__ conv_b,
    const float* __restrict__ move1,  const float* __restrict__ alpha,
    const float* __restrict__ move2,  const float* __restrict__ ln_g,
    const float* __restrict__ ln_b,   float* __restrict__ out)
{
    __shared__ float s_out[C_ALL * SPOS];        // 84 KB conv-sum intermediate
    __shared__ float s_tile[2][TILE_ELEMS + 4];  // double-buffered input tiles (+dummy slot)
    __shared__ float s_w[3 * C_ALL * 9];
    __shared__ float s_cb[3 * C_ALL], s_m1[3 * C_ALL], s_al[3 * C_ALL], s_m2[3 * C_ALL];
    __shared__ float s_bias[C_ALL], s_g[C_ALL], s_bt[C_ALL];
    __shared__ float s_mean[SPOS], s_rstd[SPOS];

    const int t  = threadIdx.x;
    const int wg = blockIdx.x;
    const int b  = wg / 28;
    const int rm = wg % 28;
    const int h0 = (rm >> 1) * 4;    // 14 row-strips of 4
    const int w0 = (rm & 1) * 28;    // 2 column-strips of 28

    // ---- stage per-channel parameters into LDS ----
    for (int i = t; i < 3 * C_ALL * 9; i += 256) s_w[i] = conv_w[i];
    for (int i = t; i < 3 * C_ALL; i += 256) {
        s_cb[i] = conv_b[i]; s_m1[i] = move1[i]; s_al[i] = alpha[i]; s_m2[i] = move2[i];
    }
    if (t < C_ALL) { s_bias[t] = bias[t]; s_g[t] = ln_g[t]; s_bt[t] = ln_b[t]; }
    __syncthreads();

    const float* xb = x + (size_t)b * C_ALL * 56 * 56;

    // ---- async-stage one 8-channel input tile chunk into LDS buffer bf ----
    auto stage_chunk = [&](int chunk, int bf) {
        const int c0 = chunk * NC;
        float* tb    = &s_tile[bf][0];
        float* dummy = &s_tile[bf][TILE_ELEMS];
        #pragma unroll
        for (int i = 0; i < ILOAD; ++i) {
            int  e    = t + i * 256;
            bool tail = (e >= TILE_ELEMS);
            int  ec   = tail ? (TILE_ELEMS - 1) : e;
            int  cl   = ec / TILE_CH;
            int  r2   = ec - cl * TILE_CH;
            int  r    = r2 / TILE_W;
            int  cc   = r2 - r * TILE_W;
            int  gh   = h0 - 5 + r;
            int  gw   = w0 - 5 + cc;
            bool valid = (gh >= 0) && (gh < 56) && (gw >= 0) && (gw < 56);
            // zero-pad emulation: invalid slots hold -bias so (x+bias) padding folds out
            if (!valid && !tail) tb[ec] = -s_bias[c0 + cl];
            int ghc = gh < 0 ? 0 : (gh > 55 ? 55 : gh);
            int gwc = gw < 0 ? 0 : (gw > 55 ? 55 : gw);
            const float* gp = xb + ((size_t)(c0 + cl) * 56 + ghc) * 56 + gwc;
            uint32_t dst = lds_off(valid ? &tb[ec] : dummy);
            // every thread issues every iteration -> exactly ILOAD per wave (deterministic ASYNCcnt)
            asm volatile("global_load_async_to_lds_b32 %0, %1, off"
                         :: "v"(dst), "v"(gp) : "memory");
        }
    };

    // ---- compute conv-sum for one chunk (wave w owns channel c0+w) ----
    auto compute_chunk = [&](int chunk) {
        const int c0   = chunk * NC;
        const int cl   = t >> 5;      // wave id = local channel
        const int lane = t & 31;
        const int c    = c0 + cl;
        const float* tb = &s_tile[chunk & 1][cl * TILE_CH];
        float wr[27];
        #pragma unroll
        for (int i = 0; i < 3; ++i)
            #pragma unroll
            for (int k = 0; k < 9; ++k) wr[i * 9 + k] = s_w[(i * C_ALL + c) * 9 + k];
        const float bs = s_bias[c];
        float base[3], al[3], m2s = 0.f;
        #pragma unroll
        for (int i = 0; i < 3; ++i) {
            float sw = 0.f;
            #pragma unroll
            for (int k = 0; k < 9; ++k) sw += wr[i * 9 + k];
            base[i] = s_cb[i * C_ALL + c] - s_m1[i * C_ALL + c] + bs * sw; // conv_b - move1 + bias*Σw
            al[i]   = s_al[i * C_ALL + c];
            m2s    += s_m2[i * C_ALL + c];
        }
        const int dil[3] = {1, 3, 5};
        #pragma unroll
        for (int j = 0; j < 4; ++j) {
            int p = lane + j * 32;
            if (p < SPOS) {
                int dy = p / 28;
                int dx = p - dy * 28;
                const float* tc = tb + (dy + 5) * TILE_W + (dx + 5);
                float s = 0.f;
                #pragma unroll
                for (int i = 0; i < 3; ++i) {
                    const int d = dil[i];
                    float acc = base[i];
                    #pragma unroll
                    for (int ky = 0; ky < 3; ++ky)
                        #pragma unroll
                        for (int kx = 0; kx < 3; ++kx)
                            acc = fmaf(tc[(ky - 1) * d * TILE_W + (kx - 1) * d],
                                       wr[i * 9 + ky * 3 + kx], acc);
                    s += (acc >= 0.f) ? acc : al[i] * acc;   // PReLU
                }
                s_out[c * SPOS + p] = s + m2s;               // + move2 (summed)
            }
        }
    };

    // ---- software pipeline: async loads of chunk k+1 overlap compute of chunk k ----
    stage_chunk(0, 0);
    for (int k = 0; k < NCHUNK; ++k) {
        if (k + 1 < NCHUNK) {
            stage_chunk(k + 1, (k + 1) & 1);
            asm volatile("s_wait_asynccnt 17" ::: "memory");  // chunk k done, k+1 in flight
        } else {
            asm volatile("s_wait_asynccnt 0" ::: "memory");
        }
        __syncthreads();
        compute_chunk(k);
        __syncthreads();
    }

    // ---- LayerNorm stats via WMMA f32 16x16x4: ones(16x4) x data(4x16) accumulates
    //      per-position channel sums (all D rows identical). Wave w owns 16 positions. ----
    {
        const int wv   = t >> 5;
        const int lane = t & 31;
        if (wv < 7) {                       // 7 waves x 16 = 112 positions
            const int p0 = wv * 16;
            const int n  = lane & 15;
            const int kh = lane >> 4;       // K half: lanes 0-15 -> K{0,1}, 16-31 -> K{2,3}
            v8f accS = {}; v8f accQ = {};
            const v2f ones = {1.f, 1.f};
            for (int c0 = 0; c0 < C_ALL; c0 += 4) {
                float b0 = s_out[(c0 + 2 * kh + 0) * SPOS + p0 + n];
                float b1 = s_out[(c0 + 2 * kh + 1) * SPOS + p0 + n];
                v2f bv = {b0, b1};
                v2f bq = {b0 * b0, b1 * b1};
                accS = __builtin_amdgcn_wmma_f32_16x16x4_f32(false, ones, false, bv,
                                                             (short)0, accS, false, false);
                accQ = __builtin_amdgcn_wmma_f32_16x16x4_f32(false, ones, false, bq,
                                                             (short)0, accQ, false, false);
            }
            if (lane < 16) {
                float mean = accS[0] * (1.f / 192.f);
                float msq  = accQ[0] * (1.f / 192.f);
                float var  = msq - mean * mean;            // biased, matches torch LN
                s_mean[p0 + n] = mean;
                s_rstd[p0 + n] = rsqrtf(var + 1e-5f);
            }
        }
    }
    __syncthreads();

    // ---- apply LN and store ----
    for (int m = t; m < C_ALL * SPOS; m += 256) {
        int c = m / SPOS;
        int p = m - c * SPOS;
        float v = (s_out[m] - s_mean[p]) * s_rstd[p] * s_g[c] + s_bt[c];
        int dy = p / 28;
        int ww = w0 + (p - dy * 28);
        int hh = h0 + dy;
        out[((size_t)(b * C_ALL + c) * 56 + hh) * 56 + ww] = v;
    }
}

extern "C" void kernel_launch(void* const* d_in, const int* in_sizes, int n_in,
                              void* d_out, int out_size, void* d_ws, size_t ws_size,
                              hipStream_t stream) {
    const float* x      = (const float*)d_in[0];
    const float* bias   = (const float*)d_in[1];
    const float* conv_w = (const float*)d_in[2];
    const float* conv_b = (const float*)d_in[3];
    const float* move1  = (const float*)d_in[4];
    const float* alpha  = (const float*)d_in[5];
    const float* move2  = (const float*)d_in[6];
    const float* ln_g   = (const float*)d_in[7];
    const float* ln_b   = (const float*)d_in[8];
    float* out = (float*)d_out;
    dim3 grid(32 * 14 * 2);   // B x (56/4) x (56/28)
    msgdc_fused<<<grid, 256, 0, stream>>>(x, bias, conv_w, conv_b,
                                          move1, alpha, move2, ln_g, ln_b, out);
}